// BatchAssignOneh_70592082477730
// MI455X (gfx1250) — compile-verified
//
#include <hip/hip_runtime.h>

typedef float v2f __attribute__((ext_vector_type(2)));
typedef float v4f __attribute__((ext_vector_type(4)));
typedef float v8f __attribute__((ext_vector_type(8)));

#define Bn    8
#define Tn    128
#define TKEEP 64
#define Cn    128
#define Fn    32
#define Kn    64
#define NELEM (Bn * TKEEP * Cn * Fn) /* 2,097,152 kept elements */

// ---------------------------------------------------------------------------
// Kernel 1: keep_idx = stable_argsort(masktime)[:t_keep] via rank counting.
// masktime[t] = mask[0, t, 0, 0] (stride C*F in the flat mask array)
// ---------------------------------------------------------------------------
__global__ void build_keep_idx_kernel(const float* __restrict__ mask,
                                      const int* __restrict__ tkeep_p,
                                      int* __restrict__ keep) {
  __shared__ float mv[Tn];
  const int t = threadIdx.x;
  mv[t] = mask[(size_t)t * (Cn * Fn)];
  __syncthreads();
  const float v = mv[t];
  int rank = 0;
  for (int u = 0; u < Tn; ++u) {
    const float uv = mv[u];
    rank += (uv < v || (uv == v && u < t)) ? 1 : 0;
  }
  if (rank < tkeep_p[0]) keep[rank] = t;
}

// ---------------------------------------------------------------------------
// Kernel 2: each wave handles 64 consecutive elements (4 rounds x 16).
// Scores via V_WMMA_F32_16X16X4_F32:  D[m,n] = x_m*(-c_n) + 0.5*c_n^2
// (rank-1 GEMM + bias; argmin-equivalent to |x-c| incl. tie order).
// B/C operands hoisted across rounds. Branchless lexicographic (s,idx)
// argmin: per-lane over 4 groups, then xor-butterfly over each 16-lane half.
// One-hot rows written with NT b128 stores (full coverage, no memset pass).
// ---------------------------------------------------------------------------
__global__ __launch_bounds__(256) void assign_onehot_wmma(
    const float* __restrict__ y, const float* __restrict__ centers,
    const int* __restrict__ keep, float* __restrict__ out) {
  const int lane = threadIdx.x & 31;
  const int q    = lane & 15;   // N coordinate / center-in-group
  const int hi   = lane >> 4;   // half-wave: M offset of 8
  const long wid = (long)blockIdx.x * (blockDim.x >> 5) + (threadIdx.x >> 5);
  const long n0  = wid << 6;    // 64 elements per wave
  if (n0 >= NELEM) return;      // wave-uniform (grid is exact anyway)

  // 64 consecutive elements stay inside one (b, r) source row: decompose once
  const int inner = (int)(n0 & (Cn * Fn - 1));
  const int r     = (int)((n0 >> 12) & (TKEEP - 1));
  const int b     = (int)(n0 >> 18);
  const int tk    = keep[r];
  const float* xsrc = y + ((size_t)b * Tn + tk) * (Cn * Fn) + inner + q;

  // Hoisted WMMA operands: B = broadcast -c_n into every K slot of column n
  // (layout-robust: A is zero for k>=1), C = bias 0.5*c_n^2 along all M.
  v2f Bop[4];
  v8f Cop[4];
#pragma unroll
  for (int g = 0; g < 4; ++g) {
    const float c = centers[g * 16 + q];
    Bop[g].x = -c;
    Bop[g].y = -c;
    const float h = 0.5f * c * c;
#pragma unroll
    for (int j = 0; j < 8; ++j) Cop[g][j] = h;
  }

  for (int tt = 0; tt < 4; ++tt) {
    // A (16x4 f32): lanes 0-15 VGPR0 = (M=lane, K=0) = x_m ; all else zero
    const float xv = xsrc[16 * tt];
    v2f A;
    A.x = (hi == 0) ? xv : 0.0f;
    A.y = 0.0f;

    float bs[8];
    int   bi[8];
#pragma unroll
    for (int g = 0; g < 4; ++g) {
      v8f D = __builtin_amdgcn_wmma_f32_16x16x4_f32(
          /*neg_a=*/false, A, /*neg_b=*/false, Bop[g],
          /*c_mod=*/(short)0, Cop[g], /*reuse_a=*/false, /*reuse_b=*/false);
#pragma unroll
      for (int j = 0; j < 8; ++j) {   // branchless combine over center groups
        const float s   = D[j];
        const int   idx = g * 16 + q; // idx ascends with g: strict '<' keeps
        if (g == 0) {                 // first occurrence on ties
          bs[j] = s;
          bi[j] = idx;
        } else {
          const int take = (int)(s < bs[j]);
          bs[j] = take ? s : bs[j];
          bi[j] = take ? idx : bi[j];
        }
      }
    }

    // branchless xor-butterfly argmin across the 16 lanes of each half-wave
#pragma unroll
    for (int j = 0; j < 8; ++j) {
      float s = bs[j];
      int   i = bi[j];
#pragma unroll
      for (int m = 1; m <= 8; m <<= 1) {
        const float s2 = __shfl_xor(s, m, 32);
        const int   i2 = __shfl_xor(i, m, 32);
        // non-short-circuit predicate -> v_cmp + v_cndmask, no EXEC branches
        const int take = (int)(s2 < s) | ((int)(s2 == s) & (int)(i2 < i));
        s = take ? s2 : s;
        i = take ? i2 : i;
      }
      bi[j] = i;  // argmin for element m = j + 8*hi (all lanes in half agree)
    }

    // one-hot rows: lane q writes float4 [4q..4q+3] of each of its 8 rows
    float* orow = out + (size_t)(n0 + 16 * tt) * Kn;
#pragma unroll
    for (int j = 0; j < 8; ++j) {
      const int m   = j + 8 * hi;
      const int idx = bi[j];
      const int k0  = 4 * q;
      v4f v;
      v.x = (idx == k0 + 0) ? 1.0f : 0.0f;
      v.y = (idx == k0 + 1) ? 1.0f : 0.0f;
      v.z = (idx == k0 + 2) ? 1.0f : 0.0f;
      v.w = (idx == k0 + 3) ? 1.0f : 0.0f;
      __builtin_nontemporal_store(v, (v4f*)(orow + (size_t)m * Kn + k0));
    }
  }
}

extern "C" void kernel_launch(void* const* d_in, const int* in_sizes, int n_in,
                              void* d_out, int out_size, void* d_ws, size_t ws_size,
                              hipStream_t stream) {
  const float* y_true  = (const float*)d_in[0];
  const float* mask    = (const float*)d_in[1];
  const float* centers = (const float*)d_in[2];
  const int*   tkeep   = (const int*)d_in[3];
  float* out  = (float*)d_out;
  int*   keep = (int*)d_ws;  // 64 ints of scratch

  build_keep_idx_kernel<<<1, Tn, 0, stream>>>(mask, tkeep, keep);

  const int waves  = NELEM / 64;  // 32768 waves, 64 elements each
  const int blocks = waves / 8;   // 256 threads = 8 waves per block
  assign_onehot_wmma<<<blocks, 256, 0, stream>>>(y_true, centers, keep, out);
}